// SpKBGATModified_59631325938130
// MI455X (gfx1250) — compile-verified
//
#include <hip/hip_runtime.h>
#include <hip/hip_bf16.h>
#include <math.h>

typedef __attribute__((ext_vector_type(2))) float v2f;
typedef __attribute__((ext_vector_type(8))) float v8f;

#define LRELU_ALPHA 0.2f

__device__ __forceinline__ float wave_sum32(float v) {
#pragma unroll
  for (int off = 16; off > 0; off >>= 1) v += __shfl_xor(v, off, 32);
  return v;
}

// ---------------------------------------------------------------------------
// fp32 WMMA NT GEMM:  C[M x N] = A[M x K] * B^T, B stored row-major [N x K]
// with row stride ldb (unit K-stride -> contiguous b64 operand loads).
// Each wave computes ONE 16-row M tile x TWO adjacent 16-col N tiles, so the
// A operand is loaded once and fed to two V_WMMA_F32_16X16X4_F32 chains:
// per 8 K-values -> 6 b64 loads + 4 WMMAs (A traffic halved vs 1 tile/wave).
// Operand layout (ISA 7.12.2): lane r=lane&15 owns row r of the tile (M for A,
// N for B); v2f halves hold K = k+koff, k+koff+1, koff=(lane>=16)*2.
// C/D: row = m0 + v + 8*(lane>=16), col = n0 + (lane&15).
// Partial tiles: row indices CLAMPED (no predicated loads); garbage lanes are
// masked at the store.  Requires K % 4 == 0 (K = 100/200 here) and 8-byte
// aligned rows (all lda/ldb/base offsets here are even).
// ---------------------------------------------------------------------------
__global__ void k_wmma_gemm_nt(const float* __restrict__ A, int lda,
                               const float* __restrict__ B, int ldb,
                               float* __restrict__ C, int ldc,
                               int M, int N, int K, int slabsN)
{
  int wave = (int)((blockIdx.x * blockDim.x + threadIdx.x) >> 5);
  int lane = threadIdx.x & 31;
  int tm = wave / slabsN;
  int ts = wave - tm * slabsN;
  if (tm >= ((M + 15) >> 4)) return;
  int m0 = tm << 4;
  int n0 = ts << 5;                              // 32-wide N slab
  int r = lane & 15;
  int koff = (lane >> 4) << 1;                   // 0 or 2
  int arow  = m0 + r;       if (arow  > M - 1) arow  = M - 1;
  int brow0 = n0 + r;       if (brow0 > N - 1) brow0 = N - 1;
  int brow1 = n0 + 16 + r;  if (brow1 > N - 1) brow1 = N - 1;
  const float* Ap  = A + (size_t)arow  * lda + koff;
  const float* Bp0 = B + (size_t)brow0 * ldb + koff;
  const float* Bp1 = B + (size_t)brow1 * ldb + koff;

  v8f acc0 = {}, acc1 = {};
  int k = 0;
  for (; k + 8 <= K; k += 8) {
    v2f a0  = *(const v2f*)(Ap  + k);
    v2f a1  = *(const v2f*)(Ap  + k + 4);
    v2f b00 = *(const v2f*)(Bp0 + k);
    v2f b01 = *(const v2f*)(Bp0 + k + 4);
    v2f b10 = *(const v2f*)(Bp1 + k);
    v2f b11 = *(const v2f*)(Bp1 + k + 4);
    acc0 = __builtin_amdgcn_wmma_f32_16x16x4_f32(false, a0, false, b00, (short)0, acc0, false, false);
    acc1 = __builtin_amdgcn_wmma_f32_16x16x4_f32(false, a0, false, b10, (short)0, acc1, false, false);
    acc0 = __builtin_amdgcn_wmma_f32_16x16x4_f32(false, a1, false, b01, (short)0, acc0, false, false);
    acc1 = __builtin_amdgcn_wmma_f32_16x16x4_f32(false, a1, false, b11, (short)0, acc1, false, false);
  }
  for (; k < K; k += 4) {
    v2f a  = *(const v2f*)(Ap  + k);
    v2f b0 = *(const v2f*)(Bp0 + k);
    v2f b1 = *(const v2f*)(Bp1 + k);
    acc0 = __builtin_amdgcn_wmma_f32_16x16x4_f32(false, a, false, b0, (short)0, acc0, false, false);
    acc1 = __builtin_amdgcn_wmma_f32_16x16x4_f32(false, a, false, b1, (short)0, acc1, false, false);
  }

  int mbase = m0 + ((lane >> 4) << 3);
  int col0 = n0 + r;
  int col1 = n0 + 16 + r;
  if (col0 < N) {
#pragma unroll
    for (int v = 0; v < 8; ++v) {
      int row = mbase + v;
      if (row < M) C[(size_t)row * ldc + col0] = acc0[v];
    }
  }
  if (col1 < N) {
#pragma unroll
    for (int v = 0; v < 8; ++v) {
      int row = mbase + v;
      if (row < M) C[(size_t)row * ldc + col1] = acc1[v];
    }
  }
}

// Small dense transpose: out[c*rowsIn + r] = in[r*colsIn + c]
__global__ void k_transpose(const float* __restrict__ in, float* __restrict__ out,
                            int rowsIn, int colsIn)
{
  int idx = blockIdx.x * blockDim.x + threadIdx.x;
  int total = rowsIn * colsIn;
  if (idx >= total) return;
  int r = idx / colsIn, c = idx - r * colsIn;
  out[(size_t)c * rowsIn + r] = in[idx];
}

// ---------------------------------------------------------------------------
// Row l2-normalize (d <= 128), one wave per row.
// ---------------------------------------------------------------------------
__global__ void k_l2norm_rows(const float* __restrict__ in, float* __restrict__ out,
                              int rows, int d)
{
  int w = (int)((blockIdx.x * blockDim.x + threadIdx.x) >> 5);
  int lane = threadIdx.x & 31;
  if (w >= rows) return;
  const float* rp = in + (size_t)w * d;
  float v[4]; float ss = 0.f;
#pragma unroll
  for (int t = 0; t < 4; ++t) {
    int j = lane + 32 * t;
    v[t] = (j < d) ? rp[j] : 0.f;
    ss += v[t] * v[t];
  }
  ss = wave_sum32(ss);
  float nrm = fmaxf(sqrtf(ss), 1e-12f);
  float* op = out + (size_t)w * d;
#pragma unroll
  for (int t = 0; t < 4; ++t) {
    int j = lane + 32 * t;
    if (j < d) op[j] = v[t] / nrm;
  }
}

// ---------------------------------------------------------------------------
// Layer-1 edge pass, both heads fused, D=100.  One wave per edge.
// edge_m = P1[src] + P2[dst] + Q[t0] (+ Q[t1] for n-hop edges)
// e = exp(-leaky_relu(dot(edge_m, a2)));  atomic segment-sum by src.
// ---------------------------------------------------------------------------
__global__ void k_edge_l1(const int* __restrict__ edge_list, const int* __restrict__ edge_type,
                          const int* __restrict__ nhop, int E1, int E,
                          const float* __restrict__ P1a, const float* __restrict__ P2a,
                          const float* __restrict__ Qa,  const float* __restrict__ a2a,
                          const float* __restrict__ P1b, const float* __restrict__ P2b,
                          const float* __restrict__ Qb,  const float* __restrict__ a2b,
                          float* __restrict__ ha, float* __restrict__ rsa,
                          float* __restrict__ hb, float* __restrict__ rsb)
{
  const int D = 100;
  int e = (int)((blockIdx.x * blockDim.x + threadIdx.x) >> 5);
  int lane = threadIdx.x & 31;
  if (e >= E) return;

  int src, dst, t0, t1;
  if (e < E1) { src = edge_list[e]; dst = edge_list[E1 + e]; t0 = edge_type[e]; t1 = -1; }
  else { int i = e - E1; src = nhop[4*i+3]; dst = nhop[4*i+0]; t0 = nhop[4*i+1]; t1 = nhop[4*i+2]; }

  const float* p1a = P1a + (size_t)src * D;
  const float* p2a = P2a + (size_t)dst * D;
  const float* p1b = P1b + (size_t)src * D;
  const float* p2b = P2b + (size_t)dst * D;
  const float* qa0 = Qa + (size_t)t0 * D;
  const float* qb0 = Qb + (size_t)t0 * D;

  float ma[4], mb[4], da = 0.f, db = 0.f;
#pragma unroll
  for (int t = 0; t < 4; ++t) {
    int j = lane + 32 * t;
    float va = 0.f, vb = 0.f;
    if (j < D) {
      va = p1a[j] + p2a[j] + qa0[j];
      vb = p1b[j] + p2b[j] + qb0[j];
      if (t1 >= 0) { va += Qa[(size_t)t1 * D + j]; vb += Qb[(size_t)t1 * D + j]; }
      da += va * a2a[j];
      db += vb * a2b[j];
    }
    ma[t] = va; mb[t] = vb;
  }
  da = wave_sum32(da); db = wave_sum32(db);
  float la = (da > 0.f) ? da : LRELU_ALPHA * da;
  float lb = (db > 0.f) ? db : LRELU_ALPHA * db;
  float ea = expf(-la), eb = expf(-lb);

  float* hra = ha + (size_t)src * D;
  float* hrb = hb + (size_t)src * D;
#pragma unroll
  for (int t = 0; t < 4; ++t) {
    int j = lane + 32 * t;
    if (j < D) { atomicAdd(&hra[j], ea * ma[t]); atomicAdd(&hrb[j], eb * mb[t]); }
  }
  if (lane == 0) { atomicAdd(&rsa[src], ea); atomicAdd(&rsb[src], eb); }
}

// ---------------------------------------------------------------------------
// Output-layer edge pass, D=200, single head. One wave per edge.
// ---------------------------------------------------------------------------
__global__ void k_edge_out(const int* __restrict__ edge_list, const int* __restrict__ edge_type,
                           const int* __restrict__ nhop, int E1, int E,
                           const float* __restrict__ P1, const float* __restrict__ P2,
                           const float* __restrict__ Q,  const float* __restrict__ a2,
                           float* __restrict__ h, float* __restrict__ rs)
{
  const int D = 200;
  int e = (int)((blockIdx.x * blockDim.x + threadIdx.x) >> 5);
  int lane = threadIdx.x & 31;
  if (e >= E) return;

  int src, dst, t0, t1;
  if (e < E1) { src = edge_list[e]; dst = edge_list[E1 + e]; t0 = edge_type[e]; t1 = -1; }
  else { int i = e - E1; src = nhop[4*i+3]; dst = nhop[4*i+0]; t0 = nhop[4*i+1]; t1 = nhop[4*i+2]; }

  const float* p1 = P1 + (size_t)src * D;
  const float* p2 = P2 + (size_t)dst * D;
  const float* q0 = Q + (size_t)t0 * D;

  float m[7], d = 0.f;
#pragma unroll
  for (int t = 0; t < 7; ++t) {
    int j = lane + 32 * t;
    float v = 0.f;
    if (j < D) {
      v = p1[j] + p2[j] + q0[j];
      if (t1 >= 0) v += Q[(size_t)t1 * D + j];
      d += v * a2[j];
    }
    m[t] = v;
  }
  d = wave_sum32(d);
  float ld = (d > 0.f) ? d : LRELU_ALPHA * d;
  float ee = expf(-ld);

  float* hr = h + (size_t)src * D;
#pragma unroll
  for (int t = 0; t < 7; ++t) {
    int j = lane + 32 * t;
    if (j < D) atomicAdd(&hr[j], ee * m[t]);
  }
  if (lane == 0) atomicAdd(&rs[src], ee);
}

// x[i, 0:100] = elu(ha/rsa), x[i, 100:200] = elu(hb/rsb)
__global__ void k_finalize_l1(const float* __restrict__ ha, const float* __restrict__ rsa,
                              const float* __restrict__ hb, const float* __restrict__ rsb,
                              float* __restrict__ x, int rows)
{
  int idx = blockIdx.x * blockDim.x + threadIdx.x;
  int total = rows * 200;
  if (idx >= total) return;
  int i = idx / 200, j = idx - i * 200;
  float hv, rv;
  if (j < 100) { hv = ha[(size_t)i * 100 + j];        rv = rsa[i]; }
  else         { hv = hb[(size_t)i * 100 + (j - 100)]; rv = rsb[i]; }
  rv = (rv == 0.f) ? 1e-12f : rv;
  float v = hv / rv;
  x[idx] = (v > 0.f) ? v : expm1f(v);
}

// x_out = elu(h / rowsum), D=200
__global__ void k_finalize_out(const float* __restrict__ h, const float* __restrict__ rs,
                               float* __restrict__ xo, int rows)
{
  int idx = blockIdx.x * blockDim.x + threadIdx.x;
  int total = rows * 200;
  if (idx >= total) return;
  int i = idx / 200;
  float rv = rs[i];
  rv = (rv == 0.f) ? 1e-12f : rv;
  float v = h[idx] / rv;
  xo[idx] = (v > 0.f) ? v : expm1f(v);
}

__global__ void k_scatter_mask(const int* __restrict__ batch_inputs, float* __restrict__ mask,
                               int npos)
{
  int i = blockIdx.x * blockDim.x + threadIdx.x;
  if (i < npos) mask[batch_inputs[3 * i + 2]] = 1.0f;
}

// out[i,:] = l2norm(G[i,:] + mask[i] * xo[i,:]), D=200, one wave per row.
__global__ void k_final_entity(const float* __restrict__ G, const float* __restrict__ xo,
                               const float* __restrict__ mask, float* __restrict__ out, int rows)
{
  const int D = 200;
  int w = (int)((blockIdx.x * blockDim.x + threadIdx.x) >> 5);
  int lane = threadIdx.x & 31;
  if (w >= rows) return;
  float mk = mask[w];
  const float* g = G + (size_t)w * D;
  const float* x = xo + (size_t)w * D;
  float v[7]; float ss = 0.f;
#pragma unroll
  for (int t = 0; t < 7; ++t) {
    int j = lane + 32 * t;
    v[t] = (j < D) ? (g[j] + mk * x[j]) : 0.f;
    ss += v[t] * v[t];
  }
  ss = wave_sum32(ss);
  float nrm = fmaxf(sqrtf(ss), 1e-12f);
  float* o = out + (size_t)w * D;
#pragma unroll
  for (int t = 0; t < 7; ++t) {
    int j = lane + 32 * t;
    if (j < D) o[j] = v[t] / nrm;
  }
}

// ---------------------------------------------------------------------------
static void gemm_nt(hipStream_t s, const float* A, int lda,
                    const float* B, int ldb,
                    float* C, int ldc, int M, int N, int K)
{
  int tilesM = (M + 15) / 16;
  int slabsN = (N + 31) / 32;            // 2 N-tiles per wave
  int waves = tilesM * slabsN;
  int grid = (waves + 7) / 8;            // 8 waves per 256-thread block
  k_wmma_gemm_nt<<<grid, 256, 0, s>>>(A, lda, B, ldb, C, ldc, M, N, K, slabsN);
}

extern "C" void kernel_launch(void* const* d_in, const int* in_sizes, int n_in,
                              void* d_out, int out_size, void* d_ws, size_t ws_size,
                              hipStream_t stream)
{
  (void)n_in; (void)out_size; (void)ws_size;

  const int*   edge_list = (const int*)  d_in[0];
  const int*   edge_type = (const int*)  d_in[1];
  const int*   batch_in  = (const int*)  d_in[2];
  const int*   nhop      = (const int*)  d_in[3];
  const float* ent_emb   = (const float*)d_in[4];
  const float* rel_emb   = (const float*)d_in[5];
  const float* W         = (const float*)d_in[6];
  const float* W_ent     = (const float*)d_in[7];
  const float* a0        = (const float*)d_in[8];
  const float* a20       = (const float*)d_in[9];
  const float* a1        = (const float*)d_in[10];
  const float* a21       = (const float*)d_in[11];
  const float* a_out     = (const float*)d_in[12];
  const float* a2_out    = (const float*)d_in[13];

  const int NN = 50000, RR = 500, D = 100, DO = 200;
  const int E1 = in_sizes[1];             // 300000
  const int E2 = in_sizes[3] / 4;         // 60000
  const int E  = E1 + E2;
  const int npos = (in_sizes[2] / 3) / 3; // BATCH/3 = 1000

  // ---- workspace layout (floats; every offset even -> 8B aligned rows) ----
  float* ws = (float*)d_ws;
  const size_t SZ_ND  = (size_t)NN * D;    // 5,000,000
  const size_t SZ_NDO = (size_t)NN * DO;   // 10,000,000
  float* ent  = ws;                        // 5M
  float* Preg = ws + SZ_ND;                // 20M : P1a,P2a,P1b,P2b -> P1o,P2o -> G
  float* P1a = Preg, *P2a = Preg + SZ_ND, *P1b = Preg + 2*SZ_ND, *P2b = Preg + 3*SZ_ND;
  float* P1o = Preg, *P2o = Preg + SZ_NDO;
  float* G   = Preg;
  float* Qa  = ws + 5*SZ_ND;               // 50k
  float* Qb  = Qa + (size_t)RR * D;        // 50k
  float* Qo  = Qb + (size_t)RR * D;        // 100k
  float* Hreg = Qo + (size_t)RR * DO;      // 10M : ha,hb -> h_o
  float* ha = Hreg, *hb = Hreg + SZ_ND, *ho = Hreg;
  float* RSreg = Hreg + SZ_NDO;            // 100k : rsa,rsb -> rs_o
  float* rsa = RSreg, *rsb = RSreg + NN, *rso = RSreg;
  float* Xreg = RSreg + 2*(size_t)NN;      // 10M : x -> x_out
  float* xbuf = Xreg, *xout = Xreg;
  float* mask = Xreg + SZ_NDO;             // 50k
  float* Wt   = mask + (size_t)NN;         // 20k : W^T      [200 x 100]
  float* Wet  = Wt + (size_t)D * DO;       // 20k : W_ent^T  [200 x 100]

  float* out_ent = (float*)d_out;                  // 50000 x 200
  float* out_rel = (float*)d_out + SZ_NDO;         // 500 x 200

  // ---- 0) zero accumulators / mask (every call: deterministic) ----
  hipMemsetAsync(Hreg,  0, SZ_NDO * sizeof(float), stream);
  hipMemsetAsync(RSreg, 0, 2 * (size_t)NN * sizeof(float), stream);
  hipMemsetAsync(mask,  0, (size_t)NN * sizeof(float), stream);

  // ---- 1) ent = l2norm(entity_embeddings); transpose W / W_entities ----
  k_l2norm_rows<<<(NN + 7) / 8, 256, 0, stream>>>(ent_emb, ent, NN, D);
  k_transpose<<<(D * DO + 255) / 256, 256, 0, stream>>>(W,     Wt,  D, DO);
  k_transpose<<<(D * DO + 255) / 256, 256, 0, stream>>>(W_ent, Wet, D, DO);

  // ---- 2) layer-1 projections (WMMA NT: a is [100 x 300] row-major) ----
  gemm_nt(stream, ent, D, a0,         3*D, P1a, D, NN, D, D);
  gemm_nt(stream, ent, D, a0 +   D,   3*D, P2a, D, NN, D, D);
  gemm_nt(stream, ent, D, a1,         3*D, P1b, D, NN, D, D);
  gemm_nt(stream, ent, D, a1 +   D,   3*D, P2b, D, NN, D, D);
  gemm_nt(stream, rel_emb, D, a0 + 2*D, 3*D, Qa, D, RR, D, D);
  gemm_nt(stream, rel_emb, D, a1 + 2*D, 3*D, Qb, D, RR, D, D);

  // ---- 3) layer-1 edge pass (both heads fused) ----
  k_edge_l1<<<(E + 7) / 8, 256, 0, stream>>>(edge_list, edge_type, nhop, E1, E,
                                             P1a, P2a, Qa, a20, P1b, P2b, Qb, a21,
                                             ha, rsa, hb, rsb);

  // ---- 4) x = [elu(ha/rsa), elu(hb/rsb)] ----
  k_finalize_l1<<<(NN * DO + 255) / 256, 256, 0, stream>>>(ha, rsa, hb, rsb, xbuf, NN);

  // ---- 5) out_relation_1 = rel_emb @ W  (NT via W^T) ----
  gemm_nt(stream, rel_emb, D, Wt, D, out_rel, DO, RR, DO, D);

  // ---- 6) Qo = out_relation_1 @ a_out[:,400:600]^T ----
  gemm_nt(stream, out_rel, DO, a_out + 2*DO, 3*DO, Qo, DO, RR, DO, DO);

  // ---- 7) output-layer projections ----
  gemm_nt(stream, xbuf, DO, a_out,       3*DO, P1o, DO, NN, DO, DO);
  gemm_nt(stream, xbuf, DO, a_out + DO,  3*DO, P2o, DO, NN, DO, DO);

  // ---- 8) re-zero accumulators for output layer ----
  hipMemsetAsync(Hreg,  0, SZ_NDO * sizeof(float), stream);
  hipMemsetAsync(RSreg, 0, 2 * (size_t)NN * sizeof(float), stream);

  // ---- 9) output-layer edge pass ----
  k_edge_out<<<(E + 7) / 8, 256, 0, stream>>>(edge_list, edge_type, nhop, E1, E,
                                              P1o, P2o, Qo, a2_out, ho, rso);

  // ---- 10) x_out = elu(ho/rso) ----
  k_finalize_out<<<(NN * DO + 255) / 256, 256, 0, stream>>>(ho, rso, xout, NN);

  // ---- 11) G = ent @ W_entities  (NT via W_ent^T) ----
  gemm_nt(stream, ent, D, Wet, D, G, DO, NN, DO, D);

  // ---- 12) mask scatter ----
  k_scatter_mask<<<(npos + 255) / 256, 256, 0, stream>>>(batch_in, mask, npos);

  // ---- 13) out_entity = l2norm(G + mask * x_out) ----
  k_final_entity<<<(NN + 7) / 8, 256, 0, stream>>>(G, xout, mask, out_ent, NN);
}